// SRP_PHAT_7507602833750
// MI455X (gfx1250) — compile-verified
//
#include <hip/hip_runtime.h>
#include <math.h>

#define NFFT   4096
#define NBINS  2049          // NFFT/2 + 1
#define KPAD   4104          // 2*2052, multiple of 8 (>= 2*NBINS, zero padded)
#define KC     512           // GEMM K-chunk staged in LDS (32 KB of B)
#define LAGS   81
#define LPAD   96            // 6 WMMA column tiles
#define MAXTAU 40
#define NMICS  8
#define TWO_PI 6.2831853071795864769f

typedef __attribute__((ext_vector_type(2))) float v2f;
typedef __attribute__((ext_vector_type(8))) float v8f;

// ---------------------------------------------------------------------------
// Kernel 1: Stockham radix-2 DIF FFT (length 4096, real input) in LDS.
// One block per (batch, mic) signal; 256 threads (8 wave32); 64 KB dynamic LDS.
// ---------------------------------------------------------------------------
__global__ __launch_bounds__(256) void fft_kernel(const float* __restrict__ sig,
                                                  float2* __restrict__ spec) {
  extern __shared__ float2 smem[];
  float2* bufA = smem;
  float2* bufB = smem + NFFT;

  const int sigId = blockIdx.x;                 // b*NMICS + m
  const float* x = sig + (size_t)sigId * NFFT;

  for (int i = threadIdx.x; i < NFFT; i += blockDim.x)
    bufA[i] = make_float2(x[i], 0.0f);
  __syncthreads();

  float2* src = bufA;
  float2* dst = bufB;
  int n = NFFT;
  int sShift = 0;
  for (int stage = 0; stage < 12; ++stage) {    // log2(4096) = 12
    const int m = n >> 1;
    const int s = 1 << sShift;
    for (int t = threadIdx.x; t < NFFT / 2; t += blockDim.x) {
      const int q = t & (s - 1);
      const int p = t >> sShift;
      float2 a = src[q + s * p];
      float2 b = src[q + s * (p + m)];
      float sn, cs;
      __sincosf(-TWO_PI * (float)p / (float)n, &sn, &cs);
      float2 sum = make_float2(a.x + b.x, a.y + b.y);
      float2 dif = make_float2(a.x - b.x, a.y - b.y);
      dst[q + s * (2 * p)]     = sum;
      dst[q + s * (2 * p + 1)] = make_float2(dif.x * cs - dif.y * sn,
                                             dif.x * sn + dif.y * cs);
    }
    __syncthreads();
    float2* tmp = src; src = dst; dst = tmp;    // after 12 swaps result in bufA
    n = m;
    ++sShift;
  }

  float2* out = spec + (size_t)sigId * NBINS;
  for (int k = threadIdx.x; k < NBINS; k += blockDim.x)
    out[k] = src[k];
}

// ---------------------------------------------------------------------------
// Kernel 2: PHAT-normalized cross spectrum -> A matrix rows (weights folded).
// ---------------------------------------------------------------------------
__global__ __launch_bounds__(128) void phat_kernel(const float2* __restrict__ spec,
                                                   const int* __restrict__ comb,
                                                   float* __restrict__ Amat,
                                                   int P) {
  const int row = blockIdx.x;                   // b*P + p
  const int b = row / P, p = row - b * P;
  const int mi = comb[p * 2 + 0];
  const int mj = comb[p * 2 + 1];
  const float2* Si = spec + (size_t)(b * NMICS + mi) * NBINS;
  const float2* Sj = spec + (size_t)(b * NMICS + mj) * NBINS;
  float* Ar = Amat + (size_t)row * KPAD;

  for (int k = threadIdx.x; k < KPAD / 2; k += blockDim.x) {
    float re = 0.0f, im = 0.0f;
    if (k < NBINS) {
      float2 si = Si[k], sj = Sj[k];
      float xr = si.x * sj.x + si.y * sj.y;     // S_i * conj(S_j)
      float xi = si.y * sj.x - si.x * sj.y;
      float inv = 1.0f / (sqrtf(xr * xr + xi * xi) + 1e-12f);
      float w = (k == 0 || k == NFFT / 2) ? (1.0f / NFFT) : (2.0f / NFFT);
      re = xr * inv * w;
      im = xi * inv * w;
    }
    Ar[2 * k + 0] = re;
    Ar[2 * k + 1] = im;
  }
}

// ---------------------------------------------------------------------------
// Kernel 3: twiddle table B[2k][lag] = cos(2*pi*k*(lag-40)/N),
//           B[2k+1][lag] = -sin(...). Exact integer-mod argument reduction.
// ---------------------------------------------------------------------------
__global__ __launch_bounds__(256) void twiddle_kernel(float* __restrict__ Bmat) {
  const int idx = blockIdx.x * blockDim.x + threadIdx.x;
  if (idx >= (KPAD / 2) * LPAD) return;
  const int k = idx / LPAD;
  const int lag = idx - k * LPAD;
  float c = 0.0f, s = 0.0f;
  if (k < NBINS) {
    const int nlag = lag - MAXTAU;
    const int ph = (k * nlag) & (NFFT - 1);     // exact mod-4096 phase
    __sincosf((float)ph * (TWO_PI / (float)NFFT), &s, &c);
  }
  Bmat[(size_t)(2 * k + 0) * LPAD + lag] = c;
  Bmat[(size_t)(2 * k + 1) * LPAD + lag] = -s;
}

// ---------------------------------------------------------------------------
// Kernel 4: C = A * B via V_WMMA_F32_16X16X4_F32.
// Block = 8 waves, each wave owns one 16x16 tile: 8 row tiles x 1 col tile.
// B column tile is staged per K-chunk into LDS with the CDNA5 async
// global->LDS path (global_load_async_to_lds_b128, ASYNCcnt) and shared by
// all 8 waves; A streams from global (L2-resident) as b64 per lane.
// ---------------------------------------------------------------------------
__global__ __launch_bounds__(256) void gemm_kernel(const float* __restrict__ A,
                                                   const float* __restrict__ Bm,
                                                   float* __restrict__ C,
                                                   int rowTiles) {
  __shared__ float Bs[KC * 16];                 // 32 KB

  const int waveId = threadIdx.x >> 5;
  const int lane = threadIdx.x & 31;
  const int rt  = blockIdx.x * 8 + waveId;      // row tile (may exceed rowTiles)
  const int rtc = (rt < rowTiles) ? rt : (rowTiles - 1);  // clamp: keep barriers uniform
  const int ct  = blockIdx.y;                   // col tile
  const int half = lane >> 4;                   // 0: K+0..1, 1: K+2..3
  const int l15 = lane & 15;
  const int koff = half * 2;

  const float* Arow = A + (size_t)(rtc * 16 + l15) * KPAD + koff;
  const float* Bcol = Bm + (size_t)ct * 16;     // + k*LPAD later

  v8f c = {0.f, 0.f, 0.f, 0.f, 0.f, 0.f, 0.f, 0.f};

  for (int kc = 0; kc < KPAD; kc += KC) {
    const int kLen = (KPAD - kc < KC) ? (KPAD - kc) : KC;

    // ---- async stage B[kc..kc+kLen) x 16 cols into LDS (16B per lane) ----
    for (int i = threadIdx.x; i < kLen * 4; i += blockDim.x) {
      const int kk = i >> 2;                    // chunk-local B row
      const int grp = i & 3;                    // 4 floats per group
      unsigned long long gaddr =
          (unsigned long long)(Bcol + (size_t)(kc + kk) * LPAD + grp * 4);
      unsigned ldsoff = (unsigned)(uintptr_t)(&Bs[kk * 16 + grp * 4]);
      asm volatile("global_load_async_to_lds_b128 %0, %1, off"
                   :: "v"(ldsoff), "v"(gaddr) : "memory");
    }
    asm volatile("s_wait_asynccnt 0x0" ::: "memory");
    __syncthreads();

    // ---- consume: WMMA over the chunk, B from LDS, A from global/L2 ----
    for (int k0 = 0; k0 < kLen; k0 += 4) {
      v2f a, b;
      a.x = Arow[kc + k0];
      a.y = Arow[kc + k0 + 1];
      b.x = Bs[(k0 + koff + 0) * 16 + l15];
      b.y = Bs[(k0 + koff + 1) * 16 + l15];
      c = __builtin_amdgcn_wmma_f32_16x16x4_f32(false, a, false, b,
                                                (short)0, c, false, false);
    }
    __syncthreads();                            // before next chunk overwrites Bs
  }

  if (rt < rowTiles) {
    const int mbase = rt * 16 + half * 8;
    const int ncol = ct * 16 + l15;
    #pragma unroll
    for (int r = 0; r < 8; ++r)
      C[(size_t)(mbase + r) * LPAD + ncol] = c[r];
  }
}

// ---------------------------------------------------------------------------
// Kernel 5: power[g] = sum_p cc[b][p][tau[g][p]]; block argmax -> atomicMax.
// tau rows are 112B (16B aligned): read as 7 x int4 (global_load_b128).
// ---------------------------------------------------------------------------
__global__ __launch_bounds__(256) void power_kernel(const float* __restrict__ C,
                                                    const int* __restrict__ tau,
                                                    unsigned long long* __restrict__ best,
                                                    int G, int P, int chunk) {
  __shared__ float cc_s[28 * LAGS];
  __shared__ unsigned long long red[256];

  const int b = blockIdx.y;
  for (int i = threadIdx.x; i < P * LAGS; i += blockDim.x) {
    const int p = i / LAGS;
    const int lag = i - p * LAGS;
    cc_s[i] = C[(size_t)(b * P + p) * LPAD + lag];
  }
  __syncthreads();

  const bool vec4 = ((P & 3) == 0);
  unsigned long long bestKey = 0ull;
  const int g0 = blockIdx.x * chunk;
  const int g1 = (g0 + chunk < G) ? (g0 + chunk) : G;
  for (int g = g0 + (int)threadIdx.x; g < g1; g += blockDim.x) {
    const int* trow = tau + (size_t)g * P;
    float acc = 0.0f;
    if (vec4) {
      const int4* t4 = (const int4*)trow;
      for (int p4 = 0; p4 < (P >> 2); ++p4) {
        int4 t = t4[p4];
        acc += cc_s[(p4 * 4 + 0) * LAGS + t.x];
        acc += cc_s[(p4 * 4 + 1) * LAGS + t.y];
        acc += cc_s[(p4 * 4 + 2) * LAGS + t.z];
        acc += cc_s[(p4 * 4 + 3) * LAGS + t.w];
      }
    } else {
      for (int p = 0; p < P; ++p)
        acc += cc_s[p * LAGS + trow[p]];
    }
    unsigned u = __float_as_uint(acc);
    u = (u & 0x80000000u) ? ~u : (u | 0x80000000u);       // order-preserving map
    unsigned long long key =
        ((unsigned long long)u << 32) | (unsigned)(0xFFFFFFFFu - (unsigned)g);
    bestKey = (key > bestKey) ? key : bestKey;
  }

  red[threadIdx.x] = bestKey;
  __syncthreads();
  for (int s = 128; s > 0; s >>= 1) {
    if ((int)threadIdx.x < s) {
      unsigned long long o = red[threadIdx.x + s];
      if (o > red[threadIdx.x]) red[threadIdx.x] = o;
    }
    __syncthreads();
  }
  if (threadIdx.x == 0) atomicMax(&best[b], red[0]);
}

__global__ void init_kernel(unsigned long long* best, int B) {
  const int i = blockIdx.x * blockDim.x + threadIdx.x;
  if (i < B) best[i] = 0ull;
}

__global__ void final_kernel(const unsigned long long* __restrict__ best,
                             const float* __restrict__ grid_x,
                             const float* __restrict__ cent,
                             float* __restrict__ out, int B) {
  const int i = blockIdx.x * blockDim.x + threadIdx.x;
  if (i < B * 3) {
    const int b = i / 3, c = i - b * 3;
    const unsigned idx = 0xFFFFFFFFu - (unsigned)(best[b] & 0xFFFFFFFFull);
    out[i] = grid_x[(size_t)idx * 3 + c] - cent[c];
  }
}

// ---------------------------------------------------------------------------
extern "C" void kernel_launch(void* const* d_in, const int* in_sizes, int n_in,
                              void* d_out, int out_size, void* d_ws, size_t ws_size,
                              hipStream_t stream) {
  const float* signal = (const float*)d_in[0];
  const float* grid_x = (const float*)d_in[1];
  const int*   tau    = (const int*)d_in[2];
  const int*   comb   = (const int*)d_in[3];
  const float* cent   = (const float*)d_in[4];
  float* out = (float*)d_out;

  const int B = in_sizes[0] / (NMICS * NFFT);   // 16
  const int G = in_sizes[1] / 3;                // ~31416 grid points
  const int P = in_sizes[3] / 2;                // 28 mic pairs
  const int R = B * P;                          // 448 GEMM rows
  const int rowTiles = (R + 15) / 16;           // 28
  const int colTiles = LPAD / 16;               // 6
  const int Rpad = rowTiles * 16;

  // workspace carve-out (~11 MB total; all L2-resident on a 192 MB L2)
  char* ws = (char*)d_ws;
  size_t off = 0;
  auto carve = [&](size_t bytes) -> char* {
    char* p = ws + off;
    off += (bytes + 255) & ~(size_t)255;
    return p;
  };
  float2* spec = (float2*)carve((size_t)B * NMICS * NBINS * sizeof(float2));
  float*  Amat = (float*)carve((size_t)Rpad * KPAD * sizeof(float));
  float*  Bmat = (float*)carve((size_t)KPAD * LPAD * sizeof(float));
  float*  Cmat = (float*)carve((size_t)Rpad * LPAD * sizeof(float));
  unsigned long long* best = (unsigned long long*)carve(B * sizeof(unsigned long long));

  init_kernel<<<1, 64, 0, stream>>>(best, B);

  fft_kernel<<<B * NMICS, 256, 2 * NFFT * sizeof(float2), stream>>>(signal, spec);

  phat_kernel<<<R, 128, 0, stream>>>(spec, comb, Amat, P);

  const int twd = (KPAD / 2) * LPAD;
  twiddle_kernel<<<(twd + 255) / 256, 256, 0, stream>>>(Bmat);

  dim3 ggrid((rowTiles + 7) / 8, colTiles);     // 4 x 6 blocks, 8 waves each
  gemm_kernel<<<ggrid, 256, 0, stream>>>(Amat, Bmat, Cmat, rowTiles);

  const int chunk = 4096;
  dim3 pgrid((G + chunk - 1) / chunk, B);
  power_kernel<<<pgrid, 256, 0, stream>>>(Cmat, tau, best, G, P, chunk);

  final_kernel<<<1, 64, 0, stream>>>(best, grid_x, cent, out, B);
}